// Matcher_15152644620426
// MI455X (gfx1250) — compile-verified
//
#include <hip/hip_runtime.h>
#include <math.h>

typedef float v2f __attribute__((ext_vector_type(2)));
typedef float v8f __attribute__((ext_vector_type(8)));

#define B_   2
#define N_   64
#define NGT_ 8
#define C_   17
#define HW_  4096
#define MT_  4          // N_/16 M-tiles

// workspace layout (in floats)
#define WS_SUMS 0       // B*NGT*C = 272 row sums
#define WS_W    512     // B*NGT*C = 272 weights valid/num_kp
#define WS_PART 1024    // B*MT*C*16*8 = 17408 partial hms tiles

// ---------- shared device math ----------
// g1 = (softplus(x)-x)*(1-p)^2 ; g2 = softplus(x)*p^2 ; g3 = -x*p^2  (p = sigmoid(x))
__device__ __forceinline__ void gfun(float x, float& g1, float& g2, float& g3) {
  float ax  = fabsf(x);
  float ez  = expf(-ax);                  // e^{-|x|}
  float sp  = log1pf(ez) + fmaxf(x, 0.f); // softplus(x), stable
  float inv = 1.f / (1.f + ez);
  float p   = (x >= 0.f) ? inv : ez * inv; // sigmoid(x)
  float omp = 1.f - p;
  g1 = (sp - x) * omp * omp;
  g2 = sp * p * p;
  g3 = -x * p * p;
}

// t-side features, scaled by weight w = valid/num_kp (w==0 kills padded cols too)
__device__ __forceinline__ void tfun(float t, float w, float& e, float& q, float& r) {
  float u  = 1.f - t;
  float u2 = u * u;
  float u4 = u2 * u2;
  e = (t == 1.0f) ? w : 0.f;
  q = u4 * w;
  r = t * u4 * w;
}

__device__ __forceinline__ v8f wmma_f32(v2f a, v2f b, v8f c) {
  // 8-arg form: (neg_a, A, neg_b, B, c_mod, C, reuse_a, reuse_b)
  return __builtin_amdgcn_wmma_f32_16x16x4_f32(false, a, false, b, (short)0, c,
                                               false, false);
}

// ---------- K1: per-(b,g,c) sum of gt heatmap row (valid detection) ----------
__global__ void k_rowsum(const float* __restrict__ gh, float* __restrict__ ws) {
  int bid = blockIdx.x;                          // (b*NGT+g)*C + c
  const float* row = gh + (size_t)bid * HW_;
  float s = 0.f;
  for (int i = threadIdx.x; i < HW_; i += 256) s += row[i];
  __shared__ float red[256];
  red[threadIdx.x] = s;
  __syncthreads();
  for (int off = 128; off > 0; off >>= 1) {
    if (threadIdx.x < off) red[threadIdx.x] += red[threadIdx.x + off];
    __syncthreads();
  }
  if (threadIdx.x == 0) ws[WS_SUMS + bid] = red[0];
}

// ---------- K2: weights w[b,g,c] = valid / max(num_valid,1) ----------
__global__ void k_weights(float* __restrict__ ws) {
  int t = threadIdx.x;                           // (b*NGT+g)
  if (t >= B_ * NGT_) return;
  int cnt = 0;
  float v[C_];
  for (int c = 0; c < C_; ++c) {
    v[c] = (ws[WS_SUMS + t * C_ + c] != 0.f) ? 1.f : 0.f;
    cnt += (v[c] != 0.f);
  }
  float nk = fmaxf((float)cnt, 1.f);
  for (int c = 0; c < C_; ++c) ws[WS_W + t * C_ + c] = v[c] / nk;
}

// ---------- K3: fused feature-gen + WMMA f32 GEMM over K = HW per (b,mtile,c) ----
// K-slot remap per 8-hw iteration: half0 lanes own hw h..h+3, half1 lanes own
// hw h+4..h+7.  WMMA step A contracts {h,h+1 | h+4,h+5}, step B contracts
// {h+2,h+3 | h+6,h+7}.  A and B operands use the identical mapping, so the sum
// over hw is unchanged, and each lane needs exactly one b128 load per operand.
__global__ void __launch_bounds__(256) k_gemm(const float* __restrict__ ph,
                                              const float* __restrict__ gh,
                                              float* __restrict__ ws) {
  int bid = blockIdx.x;
  int c   = bid % C_;
  int mt  = (bid / C_) % MT_;
  int b   = bid / (C_ * MT_);

  int lane = threadIdx.x & 31;
  int wave = threadIdx.x >> 5;
  int m    = lane & 15;       // A row (prediction within tile)
  int half = lane >> 4;       // K sub-offset selector
  int col  = lane & 15;       // B column (gt index, 8..15 are zero padding)
  int colc = col & 7;         // clamped for safe (dead) loads

  float wgt = (col < 8) ? ws[WS_W + (b * NGT_ + col) * C_ + c] : 0.f;

  const float* xrow = ph + (((size_t)(b * N_ + mt * 16 + m) * C_ + c) * HW_);
  const float* trow = gh + (((size_t)(b * NGT_ + colc) * C_ + c) * HW_);

  v8f accE = {0, 0, 0, 0, 0, 0, 0, 0};
  v8f accQ = {0, 0, 0, 0, 0, 0, 0, 0};
  v8f accR = {0, 0, 0, 0, 0, 0, 0, 0};

  const int chunk = HW_ / 8;                     // 512 hw per wave
  int h0 = wave * chunk;
  for (int h = h0; h < h0 + chunk; h += 8) {
    int o = h + 4 * half;                        // 16B aligned
    float4 xv = *(const float4*)(xrow + o);
    float4 tv = *(const float4*)(trow + o);

    float g1v[4], g2v[4], g3v[4];
    gfun(xv.x, g1v[0], g2v[0], g3v[0]);
    gfun(xv.y, g1v[1], g2v[1], g3v[1]);
    gfun(xv.z, g1v[2], g2v[2], g3v[2]);
    gfun(xv.w, g1v[3], g2v[3], g3v[3]);
    float ev[4], qv[4], rv[4];
    tfun(tv.x, wgt, ev[0], qv[0], rv[0]);
    tfun(tv.y, wgt, ev[1], qv[1], rv[1]);
    tfun(tv.z, wgt, ev[2], qv[2], rv[2]);
    tfun(tv.w, wgt, ev[3], qv[3], rv[3]);

    // step A: elements {0,1} of each half
    v2f aE0 = {g1v[0], g1v[1]}, aQ0 = {g2v[0], g2v[1]}, aR0 = {g3v[0], g3v[1]};
    v2f bE0 = {ev[0], ev[1]},   bQ0 = {qv[0], qv[1]},   bR0 = {rv[0], rv[1]};
    // step B: elements {2,3} of each half
    v2f aE1 = {g1v[2], g1v[3]}, aQ1 = {g2v[2], g2v[3]}, aR1 = {g3v[2], g3v[3]};
    v2f bE1 = {ev[2], ev[3]},   bQ1 = {qv[2], qv[3]},   bR1 = {rv[2], rv[3]};

    // interleave features so same-accumulator WMMAs are 3 apart
    accE = wmma_f32(aE0, bE0, accE);
    accQ = wmma_f32(aQ0, bQ0, accQ);
    accR = wmma_f32(aR0, bR0, accR);
    accE = wmma_f32(aE1, bE1, accE);
    accQ = wmma_f32(aQ1, bQ1, accQ);
    accR = wmma_f32(aR1, bR1, accR);
  }

  v8f acc = accE + accQ + accR;

  // reduce the 8 per-wave 16x16 partials through LDS (fixed order -> deterministic)
  __shared__ float red[8 * 256];
#pragma unroll
  for (int i = 0; i < 8; ++i) red[wave * 256 + lane * 8 + i] = acc[i];
  __syncthreads();

  int t = threadIdx.x;
  float s = 0.f;
#pragma unroll
  for (int w2 = 0; w2 < 8; ++w2) s += red[w2 * 256 + t];

  int l = t >> 3, i = t & 7;
  int M = i + 8 * (l >> 4);                      // C/D layout: vgpr i holds rows i, i+8
  int g = l & 15;
  if (g < 8)
    ws[WS_PART + (((b * MT_ + mt) * C_ + c) * 16 + M) * 8 + g] = s;
}

// ---------- K4: finalize: sum c-partials + score cost + offset cost ----------
__global__ void k_final(const float* __restrict__ ps, const float* __restrict__ po,
                        const float* __restrict__ go, const float* __restrict__ ws,
                        float* __restrict__ out) {
  int t = blockIdx.x * blockDim.x + threadIdx.x;
  if (t >= B_ * N_ * NGT_) return;
  int g = t % NGT_;
  int n = (t / NGT_) % N_;
  int b = t / (NGT_ * N_);
  int mt = n >> 4, m = n & 15;

  float hms = 0.f;
  for (int c = 0; c < C_; ++c)
    hms += ws[WS_PART + (((b * MT_ + mt) * C_ + c) * 16 + m) * 8 + g];

  // score_cost = ALPHA * bce(ps,1) * (1-sigmoid(ps))^GAMMA = 0.25 * g1(ps)
  float x = ps[b * N_ + n];
  float g1, g2, g3;
  gfun(x, g1, g2, g3);
  float score = 0.25f * g1;

  // off_cost = sum_c w * ((sig(po0)-go0)^2 + (sig(po1)-go1)^2) / 2
  float off = 0.f;
  for (int c = 0; c < C_; ++c) {
    float w  = ws[WS_W + (b * NGT_ + g) * C_ + c];
    float p0 = po[((b * N_ + n) * C_ + c) * 2 + 0];
    float p1 = po[((b * N_ + n) * C_ + c) * 2 + 1];
    float s0 = 1.f / (1.f + expf(-p0));
    float s1 = 1.f / (1.f + expf(-p1));
    float d0 = s0 - go[((b * NGT_ + g) * C_ + c) * 2 + 0];
    float d1 = s1 - go[((b * NGT_ + g) * C_ + c) * 2 + 1];
    off += w * (d0 * d0 + d1 * d1);
  }
  off *= 0.5f;

  out[t] = 2.f * hms + score + off;               // HMS_W=2, SCORE_W=1, OFF_W=1
}

extern "C" void kernel_launch(void* const* d_in, const int* in_sizes, int n_in,
                              void* d_out, int out_size, void* d_ws, size_t ws_size,
                              hipStream_t stream) {
  const float* ph = (const float*)d_in[0];   // pred_hms    (B,N,C,H,W)
  const float* gh = (const float*)d_in[1];   // gt_heatmaps (B,NGT,C,H,W)
  const float* ps = (const float*)d_in[2];   // pred_scores (B,N,1)
  const float* po = (const float*)d_in[3];   // pred_offsets(B,N,C,2)
  const float* go = (const float*)d_in[4];   // gt_offsets  (B,NGT,C,2)
  float* out = (float*)d_out;
  float* ws  = (float*)d_ws;

  k_rowsum <<<B_ * NGT_ * C_, 256, 0, stream>>>(gh, ws);
  k_weights<<<1, 32, 0, stream>>>(ws);
  k_gemm   <<<B_ * MT_ * C_, 256, 0, stream>>>(ph, gh, ws);
  k_final  <<<(B_ * N_ * NGT_ + 255) / 256, 256, 0, stream>>>(ps, po, go, ws, out);
}